// SimplePoxelGCN_89034672046773
// MI455X (gfx1250) — compile-verified
//
#include <hip/hip_runtime.h>
#include <hip/hip_bf16.h>

#define HDIM 128
#define LDSW 132   // padded LDS stride (floats) to avoid bank conflicts

typedef float v2f __attribute__((ext_vector_type(2)));
typedef float v8f __attribute__((ext_vector_type(8)));

// ---- hardware f32 atomic add (global_atomic_add_f32) ----
__device__ __forceinline__ void atomAddF(float* p, float v) {
    unsafeAtomicAdd(p, v);
}

// ---- generic 32-bit fill ----
__global__ void fill_u32_kernel(unsigned* __restrict__ p, unsigned v, long long n) {
    long long i = (long long)blockIdx.x * blockDim.x + threadIdx.x;
    if (i < n) p[i] = v;
}

// ---- degree: deg[dst] += w over edges, +1 for self loops ----
__global__ void deg_kernel(const int* __restrict__ ei, const float* __restrict__ w,
                           float* __restrict__ deg, int E, int N) {
    int i = blockIdx.x * blockDim.x + threadIdx.x;
    if (i < E) {
        atomAddF(&deg[ei[E + i]], w[i]);          // dst row of edge_index
    } else if (i < E + N) {
        atomAddF(&deg[i - E], 1.0f);              // self loop weight 1
    }
}

// ---- dinv = deg>0 ? rsqrt(deg) : 0 (in place), selfnorm = dinv^2 ----
__global__ void dinv_kernel(float* __restrict__ deg, float* __restrict__ selfn, int N) {
    int i = blockIdx.x * blockDim.x + threadIdx.x;
    if (i >= N) return;
    float d = deg[i];
    float r = (d > 0.0f) ? rsqrtf(d) : 0.0f;
    deg[i]   = r;
    selfn[i] = r * r;
}

// ---- per-edge norm = dinv[src] * w * dinv[dst] ----
__global__ void norm_kernel(const int* __restrict__ ei, const float* __restrict__ w,
                            const float* __restrict__ dinv, float* __restrict__ norm, int E) {
    int e = blockIdx.x * blockDim.x + threadIdx.x;
    if (e >= E) return;
    norm[e] = dinv[ei[e]] * w[e] * dinv[ei[E + e]];
}

// ---- SpMM scatter: one wave32 per edge (or self loop), 4 floats/lane ----
__global__ __launch_bounds__(256)
void scatter_kernel(const float* __restrict__ h, const int* __restrict__ ei,
                    const float* __restrict__ norm, const float* __restrict__ selfn,
                    float* __restrict__ agg, int N, int E) {
    int wave = blockIdx.x * (blockDim.x >> 5) + (threadIdx.x >> 5);
    int lane = threadIdx.x & 31;
    int s, d; float w;
    if (wave < E) {
        s = ei[wave]; d = ei[E + wave]; w = norm[wave];
    } else {
        int n = wave - E;
        if (n >= N) return;
        s = n; d = n; w = selfn[n];
    }
    const float4 hv = ((const float4*)(h + (size_t)s * HDIM))[lane];
    float* dst = agg + (size_t)d * HDIM + lane * 4;
    atomAddF(dst + 0, hv.x * w);
    atomAddF(dst + 1, hv.y * w);
    atomAddF(dst + 2, hv.z * w);
    atomAddF(dst + 3, hv.w * w);
}

// ---- fused mix + GEMM (+bias) (+ReLU) with V_WMMA_F32_16X16X4_F32 ----
// out[m, :] = act( ((1-alpha)*A[m,:] + alpha*X0[m,:]) @ W + bias )
// grid.x = ceil(N/16), block = 256 (8 waves); wave w owns output cols [16w,16w+16)
__global__ __launch_bounds__(256)
void gemm_kernel(const float* __restrict__ A, const float* __restrict__ X0,
                 const float* __restrict__ W, const float* __restrict__ bias,
                 float* __restrict__ out, int N, float alpha, int do_relu) {
    __shared__ float tile[16 * LDSW];
    const int tid = threadIdx.x;
    const int m0  = blockIdx.x * 16;
    const float oma = 1.0f - alpha;

    // stage 16x128 mix tile into LDS (rows past N zero-filled)
    #pragma unroll
    for (int i = 0; i < 8; ++i) {
        int idx = tid + i * 256;
        int r = idx >> 7, c = idx & 127;
        int node = m0 + r;
        float v = 0.0f;
        if (node < N) {
            size_t o = (size_t)node * HDIM + c;
            v = oma * A[o] + alpha * X0[o];
        }
        tile[r * LDSW + c] = v;
    }
    __syncthreads();

    const int lane   = tid & 31;
    const int lanelo = lane & 15;
    const int half   = lane >> 4;
    const int n0     = (tid >> 5) * 16;

    v8f acc = {};
    for (int k0 = 0; k0 < HDIM; k0 += 4) {
        // A fragment (16x4 f32): lanes 0-15 hold K=k0,k0+1; lanes 16-31 hold K=k0+2,k0+3
        int ka = k0 + 2 * half;
        v2f a, b;
        a.x = tile[lanelo * LDSW + ka];
        a.y = tile[lanelo * LDSW + ka + 1];
        // B fragment (4x16 f32): VGPR g holds rows K=k0+g (lanes 0-15), K=k0+g+2 (lanes 16-31)
        b.x = W[(k0 + 2 * half)     * HDIM + n0 + lanelo];
        b.y = W[(k0 + 2 * half + 1) * HDIM + n0 + lanelo];
        acc = __builtin_amdgcn_wmma_f32_16x16x4_f32(
                  false, a, false, b, (short)0, acc, false, false);
    }

    // D layout: VGPR g -> row m0+g (lanes 0-15) / m0+g+8 (lanes 16-31), col n0+lanelo
    const int col = n0 + lanelo;
    const float bv = bias ? bias[col] : 0.0f;
    #pragma unroll
    for (int g = 0; g < 8; ++g) {
        int row = m0 + g + half * 8;
        if (row < N) {
            float v = acc[g] + bv;
            if (do_relu) v = fmaxf(v, 0.0f);
            out[(size_t)row * HDIM + col] = v;
        }
    }
}

// ---- orderable-uint encoding for float atomicMax ----
__device__ __forceinline__ unsigned f2ord(float f) {
    unsigned u = __float_as_uint(f);
    return (u & 0x80000000u) ? ~u : (u | 0x80000000u);
}
__device__ __forceinline__ float ord2f(unsigned k) {
    unsigned u = (k & 0x80000000u) ? (k & 0x7FFFFFFFu) : ~k;
    return __uint_as_float(u);
}

// ---- gate = h @ gate_w + gate_b ; segment max via atomicMax ----
__global__ void gate_kernel(const float* __restrict__ h, const float* __restrict__ gw,
                            const float* __restrict__ gb, const int* __restrict__ batch,
                            float* __restrict__ gate, unsigned* __restrict__ gmaxu, int N) {
    int n = blockIdx.x * blockDim.x + threadIdx.x;
    if (n >= N) return;
    const float4* hr = (const float4*)(h + (size_t)n * HDIM);
    const float4* wr = (const float4*)gw;
    float acc = 0.0f;
    #pragma unroll 4
    for (int i = 0; i < HDIM / 4; ++i) {
        float4 a = hr[i], b = wr[i];
        acc += a.x * b.x + a.y * b.y + a.z * b.z + a.w * b.w;
    }
    acc += gb[0];
    gate[n] = acc;
    atomicMax(&gmaxu[batch[n]], f2ord(acc));
}

// ---- eg = exp(gate - gmax[batch]); gsum += eg (gate overwritten with eg) ----
__global__ void eg_kernel(float* __restrict__ gate, const unsigned* __restrict__ gmaxu,
                          const int* __restrict__ batch, float* __restrict__ gsum, int N) {
    int n = blockIdx.x * blockDim.x + threadIdx.x;
    if (n >= N) return;
    int b = batch[n];
    float e = expf(gate[n] - ord2f(gmaxu[b]));
    gate[n] = e;
    atomAddF(&gsum[b], e);
}

// ---- out[batch[n], :] += (eg[n]/gsum[batch[n]]) * t[n, :] ----
__global__ void outagg_kernel(const float* __restrict__ t, const float* __restrict__ eg,
                              const float* __restrict__ gsum, const int* __restrict__ batch,
                              float* __restrict__ out, int N) {
    long long idx = (long long)blockIdx.x * blockDim.x + threadIdx.x;  // N * 32
    int n  = (int)(idx >> 5);
    int kc = ((int)idx & 31) * 4;
    if (n >= N) return;
    int b = batch[n];
    float a = eg[n] / gsum[b];
    float4 tv = *(const float4*)(t + (size_t)n * HDIM + kc);
    float* o = out + (size_t)b * HDIM + kc;
    atomAddF(o + 0, a * tv.x);
    atomAddF(o + 1, a * tv.y);
    atomAddF(o + 2, a * tv.z);
    atomAddF(o + 3, a * tv.w);
}

static inline unsigned cdiv(long long a, long long b) { return (unsigned)((a + b - 1) / b); }

extern "C" void kernel_launch(void* const* d_in, const int* in_sizes, int n_in,
                              void* d_out, int out_size, void* d_ws, size_t ws_size,
                              hipStream_t stream) {
    const float* x      = (const float*)d_in[0];
    const float* ew     = (const float*)d_in[1];
    const float* W1     = (const float*)d_in[2];
    const float* W2     = (const float*)d_in[3];
    const float* W3     = (const float*)d_in[4];
    const float* gate_w = (const float*)d_in[5];
    const float* gate_b = (const float*)d_in[6];
    const float* nn_w   = (const float*)d_in[7];
    const float* nn_b   = (const float*)d_in[8];
    const int*   ei     = (const int*)d_in[9];    // [2, E] flat: src row then dst row
    const int*   batch  = (const int*)d_in[10];

    const int N = in_sizes[10];
    const int E = in_sizes[1];
    const int G = out_size / HDIM;
    const size_t NH = (size_t)N * HDIM;

    // workspace layout
    float*    agg   = (float*)d_ws;        // [N,H]  (also reused for t)
    float*    hbuf  = agg   + NH;          // [N,H]
    float*    normv = hbuf  + NH;          // [E]
    float*    dinv  = normv + E;           // [N] (deg, then dinv in place)
    float*    selfn = dinv  + N;           // [N]
    float*    gate  = selfn + N;           // [N] (then eg in place)
    float*    gsum  = gate  + N;           // [G]
    unsigned* gmaxu = (unsigned*)(gsum + G); // [G]

    const int T = 256;
    const unsigned waveBlocks = cdiv((long long)E + N, 8); // 8 waves/block, 1 wave/edge

    // ---- normalization ----
    fill_u32_kernel<<<cdiv(N, T), T, 0, stream>>>((unsigned*)dinv, 0u, N);
    deg_kernel<<<cdiv((long long)E + N, T), T, 0, stream>>>(ei, ew, dinv, E, N);
    dinv_kernel<<<cdiv(N, T), T, 0, stream>>>(dinv, selfn, N);
    norm_kernel<<<cdiv(E, T), T, 0, stream>>>(ei, ew, dinv, normv, E);

    // ---- layer 1: agg(x) -> hbuf = relu(mix @ W1), alpha=0.2 ----
    fill_u32_kernel<<<cdiv((long long)NH, T), T, 0, stream>>>((unsigned*)agg, 0u, (long long)NH);
    scatter_kernel<<<waveBlocks, T, 0, stream>>>(x, ei, normv, selfn, agg, N, E);
    gemm_kernel<<<cdiv(N, 16), T, 0, stream>>>(agg, x, W1, nullptr, hbuf, N, 0.2f, 1);

    // ---- layer 2 ----
    fill_u32_kernel<<<cdiv((long long)NH, T), T, 0, stream>>>((unsigned*)agg, 0u, (long long)NH);
    scatter_kernel<<<waveBlocks, T, 0, stream>>>(hbuf, ei, normv, selfn, agg, N, E);
    gemm_kernel<<<cdiv(N, 16), T, 0, stream>>>(agg, x, W2, nullptr, hbuf, N, 0.2f, 1);

    // ---- layer 3 (alpha=0.4) ----
    fill_u32_kernel<<<cdiv((long long)NH, T), T, 0, stream>>>((unsigned*)agg, 0u, (long long)NH);
    scatter_kernel<<<waveBlocks, T, 0, stream>>>(hbuf, ei, normv, selfn, agg, N, E);
    gemm_kernel<<<cdiv(N, 16), T, 0, stream>>>(agg, x, W3, nullptr, hbuf, N, 0.4f, 1);

    // ---- attentional aggregation ----
    fill_u32_kernel<<<cdiv(G, T), T, 0, stream>>>(gmaxu, 0u, G);
    fill_u32_kernel<<<cdiv(G, T), T, 0, stream>>>((unsigned*)gsum, 0u, G);
    fill_u32_kernel<<<cdiv(G * HDIM, T), T, 0, stream>>>((unsigned*)d_out, 0u, G * HDIM);

    gate_kernel<<<cdiv(N, T), T, 0, stream>>>(hbuf, gate_w, gate_b, batch, gate, gmaxu, N);
    // t = h @ nn_w + nn_b  (alpha=0 -> plain GEMM; no relu), written into agg buffer
    gemm_kernel<<<cdiv(N, 16), T, 0, stream>>>(hbuf, hbuf, nn_w, nn_b, agg, N, 0.0f, 0);
    eg_kernel<<<cdiv(N, T), T, 0, stream>>>(gate, gmaxu, batch, gsum, N);
    outagg_kernel<<<cdiv((long long)N * 32, T), T, 0, stream>>>(agg, gate, gsum, batch,
                                                                (float*)d_out, N);
}